// StateSpaceModel_77386720739553
// MI455X (gfx1250) — compile-verified
//
#include <hip/hip_runtime.h>
#include <hip/hip_bf16.h>

typedef float v2f __attribute__((ext_vector_type(2)));
typedef float v8f __attribute__((ext_vector_type(8)));
typedef int   v4i __attribute__((ext_vector_type(4)));

#if defined(__AMDGCN__) && __has_builtin(__builtin_amdgcn_global_load_async_to_lds_b128)
#define ATHENA_ASYNC_LDS 1
#else
#define ATHENA_ASYNC_LDS 0
#endif

// global_load_async_to_lds_b128 takes (int4 AS1*, int4 AS3*, imm off, imm cpol)
#define AS1V4(p) ((__attribute__((address_space(1))) v4i*)(p))
#define AS3V4(p) ((__attribute__((address_space(3))) v4i*)(p))

// ---------------------------------------------------------------------------
// f32 WMMA 16x16x4 wrapper (D = A*B + C), builtin preferred, asm fallback.
// ---------------------------------------------------------------------------
__device__ __forceinline__ v8f wmma_f32_4(v2f a, v2f b, v8f c) {
#if defined(__AMDGCN__) && __has_builtin(__builtin_amdgcn_wmma_f32_16x16x4_f32)
  return __builtin_amdgcn_wmma_f32_16x16x4_f32(
      /*neg_a=*/false, a, /*neg_b=*/false, b,
      /*c_mod=*/(short)0, c, /*reuse_a=*/false, /*reuse_b=*/false);
#else
  v8f d = c;
  asm volatile("v_wmma_f32_16x16x4_f32 %0, %1, %2, %0"
               : "+v"(d) : "v"(a), "v"(b));
  return d;
#endif
}

// ---------------------------------------------------------------------------
// Async global -> LDS tile copy of 4096 floats (16 KB), 256 threads.
// Uses GLOBAL_LOAD_ASYNC_TO_LDS_B128 (ASYNCcnt) when available; the caller
// must execute wait_async_all() + __syncthreads() before consuming the tile.
// ---------------------------------------------------------------------------
__device__ __forceinline__ void async_tile_load4k(const float* __restrict__ g,
                                                  float* __restrict__ l, int tid) {
#if ATHENA_ASYNC_LDS
#pragma unroll
  for (int i = 0; i < 4; ++i) {
    const float* gp = g + (size_t)(i * 256 + tid) * 4;
    float* lp = l + (size_t)(i * 256 + tid) * 4;
    __builtin_amdgcn_global_load_async_to_lds_b128(AS1V4(gp), AS3V4(lp), 0, 0);
  }
#else
  for (int idx = tid; idx < 1024; idx += 256)
    ((float4*)l)[idx] = ((const float4*)g)[idx];
#endif
}

__device__ __forceinline__ void wait_async_all() {
#if defined(__AMDGCN__) && __has_builtin(__builtin_amdgcn_s_wait_asynccnt)
  __builtin_amdgcn_s_wait_asynccnt(0);
#elif ATHENA_ASYNC_LDS
  asm volatile("s_wait_asynccnt 0" ::: "memory");
#endif
}

// ---------------------------------------------------------------------------
// 64x64x64 LDS matmul: D = opA(A) * opB(B).  256 threads (8 waves), each wave
// computes two 16x16 tiles; 16 K-steps of V_WMMA_F32_16X16X4_F32.
// TA: read A transposed (logical A[m][k] = stored[k*64+m])
// TB: read B transposed (logical B[k][n] = stored[n*64+k])
// Caller handles __syncthreads() before (srcs ready) and after (dst ready).
// ---------------------------------------------------------------------------
template <bool TA, bool TB>
__device__ __forceinline__ void mm64(const float* __restrict__ A,
                                     const float* __restrict__ B,
                                     float* __restrict__ D, int tid) {
  const int lane = tid & 31;
  const int wave = tid >> 5;
  const int half = lane >> 4;   // 0: lanes 0-15, 1: lanes 16-31
  const int lr   = lane & 15;
#pragma unroll
  for (int sub = 0; sub < 2; ++sub) {
    const int tile = wave * 2 + sub;        // 0..15
    const int tm = (tile >> 2) * 16;
    const int tn = (tile & 3) * 16;
    v8f acc = {};
#pragma unroll
    for (int kk = 0; kk < 64; kk += 4) {
      const int k0 = kk + half * 2;         // this half-wave's K pair
      v2f a, b;
      a.x = TA ? A[(k0    ) * 64 + tm + lr] : A[(tm + lr) * 64 + k0    ];
      a.y = TA ? A[(k0 + 1) * 64 + tm + lr] : A[(tm + lr) * 64 + k0 + 1];
      b.x = TB ? B[(tn + lr) * 64 + k0    ] : B[(k0    ) * 64 + tn + lr];
      b.y = TB ? B[(tn + lr) * 64 + k0 + 1] : B[(k0 + 1) * 64 + tn + lr];
      acc = wmma_f32_4(a, b, acc);
    }
#pragma unroll
    for (int p = 0; p < 8; ++p) {
      const int m = tm + p + half * 8;      // C/D layout: VGPR p -> M=p / p+8
      const int n = tn + lr;
      D[m * 64 + n] = acc[p];
    }
  }
}

// ---------------------------------------------------------------------------
// In-place 64x64 Gauss-Jordan inversion (no pivoting; S is SPD here).
// 256 threads. colk: 64-float scratch; pivs: 1-float scratch.
// Ends with a __syncthreads().
// ---------------------------------------------------------------------------
__device__ void inv64_inplace(float* __restrict__ S, float* __restrict__ colk,
                              float* __restrict__ pivs, int tid) {
  for (int k = 0; k < 64; ++k) {
    if (tid == 0) pivs[0] = 1.0f / S[k * 64 + k];
    if (tid >= 64 && tid < 128) { int i = tid - 64; colk[i] = S[i * 64 + k]; }
    __syncthreads();
    const float p = pivs[0];
    if (tid < 64 && tid != k) S[k * 64 + tid] *= p;   // scale row k (j != k)
    __syncthreads();
    for (int idx = tid; idx < 4096; idx += 256) {
      const int i = idx >> 6, j = idx & 63;
      if (i == k) continue;
      if (j == k) S[idx] = -colk[i] * p;
      else        S[idx] -= colk[i] * S[k * 64 + j];
    }
    if (tid == 0) S[k * 64 + k] = p;
    __syncthreads();
  }
}

// ---------------------------------------------------------------------------
// Kernel 0: Q = Lq Lq^T + 1e-3 I, R = Lr Lr^T + 1e-3 I  (both 64x64)
// ---------------------------------------------------------------------------
__global__ void qr_kernel(const float* __restrict__ Lq,
                          const float* __restrict__ Lr,
                          float* __restrict__ Q, float* __restrict__ R) {
  const int tid = blockIdx.x * blockDim.x + threadIdx.x;
  if (tid < 4096) {
    const int i = tid >> 6, j = tid & 63;
    float s = 0.f;
    for (int k = 0; k < 64; ++k) s += Lq[i * 64 + k] * Lq[j * 64 + k];
    Q[tid] = s + ((i == j) ? 1e-3f : 0.f);
  } else if (tid < 8192) {
    const int e = tid - 4096;
    const int i = e >> 6, j = e & 63;
    float s = 0.f;
    for (int k = 0; k < 64; ++k) s += Lr[i * 64 + k] * Lr[j * 64 + k];
    R[e] = s + ((i == j) ? 1e-3f : 0.f);
  }
}

__device__ __forceinline__ float sigmoidf(float x) {
  return 1.0f / (1.0f + __expf(-x));
}

// ---------------------------------------------------------------------------
// Kernel 1: single-layer LSTM (gate order i,f,g,o) + softmax over K=32.
// One block per batch element, 128 threads (one per gate row).
// ---------------------------------------------------------------------------
__global__ void lstm_kernel(const float* __restrict__ x,
                            const float* __restrict__ W_ih,
                            const float* __restrict__ W_hh,
                            const float* __restrict__ b_ih,
                            const float* __restrict__ b_hh,
                            float* __restrict__ wout, int T, int B) {
  __shared__ float h[32], c[32], gates[128], xs[64], red[32];
  const int b = blockIdx.x;
  const int j = threadIdx.x;           // 0..127
  if (j < 32) { h[j] = 0.f; c[j] = 0.f; }
  __syncthreads();
  for (int t = 0; t < T; ++t) {
    if (j < 64) xs[j] = x[((size_t)t * B + b) * 64 + j];
    __syncthreads();
    float g = b_ih[j] + b_hh[j];
    for (int i = 0; i < 64; ++i) g += xs[i] * W_ih[j * 64 + i];
    for (int k = 0; k < 32; ++k) g += h[k] * W_hh[j * 32 + k];
    gates[j] = g;
    __syncthreads();
    if (j < 32) {
      const float ig = sigmoidf(gates[j]);
      const float fg = sigmoidf(gates[32 + j]);
      const float gg = tanhf(gates[64 + j]);
      const float og = sigmoidf(gates[96 + j]);
      const float cn = fg * c[j] + ig * gg;
      c[j] = cn;
      h[j] = og * tanhf(cn);
    }
    __syncthreads();
    if (j < 32) {                       // softmax over K
      float m = -1e30f;
      for (int k = 0; k < 32; ++k) m = fmaxf(m, h[k]);
      red[j] = __expf(h[j] - m);
    }
    __syncthreads();
    if (j < 32) {
      float s = 0.f;
      for (int k = 0; k < 32; ++k) s += red[k];
      wout[((size_t)t * B + b) * 32 + j] = red[j] / s;
    }
    __syncthreads();
  }
}

// ---------------------------------------------------------------------------
// Kernel 2: mat_As / mat_Cs = w[16384,32] @ K-bank[32,4096]  (f32 WMMA GEMM)
// grid (TB/64, 4096/64, 2); block 256; 16KB LDS.
// ---------------------------------------------------------------------------
__global__ void mix_gemm_kernel(const float* __restrict__ w,
                                const float* __restrict__ Ak,
                                const float* __restrict__ Ck,
                                float* __restrict__ Asb,
                                float* __restrict__ Csb) {
  extern __shared__ float sm[];
  float* wt = sm;          // 64x32
  float* kt = sm + 2048;   // 32x64
  const float* Kmat = blockIdx.z ? Ck : Ak;
  float* Dmat = blockIdx.z ? Csb : Asb;
  const int m0 = blockIdx.x * 64;
  const int n0 = blockIdx.y * 64;
  const int tid = threadIdx.x;
  for (int idx = tid; idx < 2048; idx += 256) {
    const int r = idx >> 5, cc = idx & 31;
    wt[idx] = w[(size_t)(m0 + r) * 32 + cc];
    const int kr = idx >> 6, kc = idx & 63;
    kt[idx] = Kmat[(size_t)kr * 4096 + n0 + kc];
  }
  __syncthreads();
  const int lane = tid & 31, wave = tid >> 5;
  const int half = lane >> 4, lr = lane & 15;
#pragma unroll
  for (int sub = 0; sub < 2; ++sub) {
    const int tile = wave * 2 + sub;
    const int tm = (tile >> 2) * 16, tn = (tile & 3) * 16;
    v8f acc = {};
#pragma unroll
    for (int kk = 0; kk < 32; kk += 4) {
      const int k0 = kk + half * 2;
      v2f a, b;
      a.x = wt[(tm + lr) * 32 + k0];
      a.y = wt[(tm + lr) * 32 + k0 + 1];
      b.x = kt[(k0    ) * 64 + tn + lr];
      b.y = kt[(k0 + 1) * 64 + tn + lr];
      acc = wmma_f32_4(a, b, acc);
    }
#pragma unroll
    for (int p = 0; p < 8; ++p) {
      const int m = m0 + tm + p + half * 8;
      const int n = n0 + tn + lr;
      Dmat[(size_t)m * 4096 + n] = acc[p];
    }
  }
}

// ---------------------------------------------------------------------------
// Kernel 3: Kalman filter forward scan. One block (256 thr) per batch elem.
// A_t/C_t tiles are double-buffered: step t prefetches step t+1's 32 KB via
// GLOBAL_LOAD_ASYNC_TO_LDS_B128 and overlaps the DMA with 6 WMMA matmuls +
// the Gauss-Jordan inverse. LDS: 8 x 16 KB + vectors (~130 KB dynamic).
// ---------------------------------------------------------------------------
__global__ void filter_kernel(const float* __restrict__ as_,
                              const float* __restrict__ Asg,
                              const float* __restrict__ Csg,
                              const float* __restrict__ Qg,
                              const float* __restrict__ Rg,
                              const float* __restrict__ init_mean,
                              const float* __restrict__ init_cov,
                              float* __restrict__ fmeans,
                              float* __restrict__ fcovs,
                              float* __restrict__ mpred0,
                              int T, int B) {
  extern __shared__ float sm[];
  float* P   = sm;                 // cov_p
  float* Ab0 = sm + 1 * 4096;      // A_t (buffer 0)
  float* Cb0 = sm + 2 * 4096;      // C_t (buffer 0)
  float* Ab1 = sm + 3 * 4096;      // A_t (buffer 1)
  float* Cb1 = sm + 4 * 4096;      // C_t (buffer 1)
  float* T1  = sm + 5 * 4096;
  float* T2  = sm + 6 * 4096;
  float* Sb  = sm + 7 * 4096;      // S -> S^-1, later scratch
  float* meanp = sm + 8 * 4096;    // 64
  float* meant = meanp + 64;       // 64
  float* innov = meant + 64;       // 64
  float* colk  = innov + 64;       // 64
  float* pivs  = colk + 64;        // 1

  const int b = blockIdx.x;
  const int tid = threadIdx.x;

  // prologue: start DMA of t=0 tiles, init carry state
  async_tile_load4k(Asg + ((size_t)0 * B + b) * 4096, Ab0, tid);
  async_tile_load4k(Csg + ((size_t)0 * B + b) * 4096, Cb0, tid);
  for (int idx = tid; idx < 4096; idx += 256) P[idx] = init_cov[idx];
  if (tid < 64) meanp[tid] = init_mean[tid];
  __syncthreads();

  for (int t = 0; t < T; ++t) {
    const size_t tb = (size_t)t * B + b;
    float* Ab = (t & 1) ? Ab1 : Ab0;
    float* Cb = (t & 1) ? Cb1 : Cb0;

    wait_async_all();               // this step's tiles landed in LDS
    __syncthreads();                // ... and are visible to all waves
    if (t + 1 < T) {                // prefetch next step into the dead buffer
      const size_t tb1 = (size_t)(t + 1) * B + b;
      float* An = (t & 1) ? Ab0 : Ab1;
      float* Cn = (t & 1) ? Cb0 : Cb1;
      async_tile_load4k(Asg + tb1 * 4096, An, tid);
      async_tile_load4k(Csg + tb1 * 4096, Cn, tid);
    }

    mm64<false, false>(Cb, P, T1, tid);        // T1 = C P
    __syncthreads();
    mm64<false, true>(T1, Cb, Sb, tid);        // Sb = C P C^T
    __syncthreads();
    for (int idx = tid; idx < 4096; idx += 256) Sb[idx] += Rg[idx];  // + R
    __syncthreads();
    inv64_inplace(Sb, colk, pivs, tid);        // Sb = S^-1

    mm64<true, false>(T1, Sb, T2, tid);        // Kt = (C P)^T S^-1 = P C^T S^-1
    __syncthreads();

    if (tid < 64) {                            // innov = a_t - C mean_p
      float s = 0.f;
      for (int j = 0; j < 64; ++j) s += Cb[tid * 64 + j] * meanp[j];
      innov[tid] = as_[tb * 64 + tid] - s;
    }
    __syncthreads();
    if (tid < 64) {                            // mean_t = mean_p + Kt innov
      float s = 0.f;
      for (int j = 0; j < 64; ++j) s += T2[tid * 64 + j] * innov[j];
      meant[tid] = meanp[tid] + s;
      fmeans[tb * 64 + tid] = meant[tid];
    }
    __syncthreads();

    mm64<false, false>(T2, T1, Sb, tid);       // Sb = Kt C P
    __syncthreads();
    for (int idx = tid; idx < 4096; idx += 256) {   // cov_t = sym(P - Kt C P)
      const int i = idx >> 6, j = idx & 63;
      const float v = 0.5f * ((P[idx] - Sb[idx]) + (P[j * 64 + i] - Sb[j * 64 + i]));
      Cb[idx] = v;                             // reuse C buffer for cov_t
      fcovs[tb * 4096 + idx] = v;
    }
    __syncthreads();

    float mn = 0.f;                            // mean_n = A mean_t
    if (tid < 64) {
      for (int j = 0; j < 64; ++j) mn += Ab[tid * 64 + j] * meant[j];
    }
    mm64<false, false>(Ab, Cb, T1, tid);       // T1 = A cov_t
    __syncthreads();
    mm64<false, true>(T1, Ab, T2, tid);        // T2 = A cov_t A^T
    __syncthreads();
    for (int idx = tid; idx < 4096; idx += 256) {   // P = sym(T2) + Q
      const int i = idx >> 6, j = idx & 63;
      P[idx] = 0.5f * (T2[idx] + T2[j * 64 + i]) + Qg[idx];
    }
    if (tid < 64) {
      meanp[tid] = mn;
      if (t == 0) mpred0[(size_t)b * 64 + tid] = mn;   // f_nmeans[0]
    }
    __syncthreads();
  }
}

// ---------------------------------------------------------------------------
// Kernel 4: RTS smoother (reverse scan). One block (256 thr) per batch elem.
// Same double-buffered async prefetch of (f_covs[t], mat_As[t]); recomputes
// Pn = sym(A P A^T) + Q with the filter's exact instruction sequence.
// LDS: 9 x 16 KB + vectors (~146 KB dynamic).
// ---------------------------------------------------------------------------
__global__ void smoother_kernel(const float* __restrict__ Asg,
                                const float* __restrict__ Qg,
                                const float* __restrict__ fmeans,
                                const float* __restrict__ fcovs,
                                const float* __restrict__ mpred0g,
                                float* __restrict__ out,
                                int T, int B) {
  extern __shared__ float sm[];
  float* Pb0  = sm;                // f_covs[t] (buffer 0)
  float* Ab0  = sm + 1 * 4096;     // mat_As[t] (buffer 0)
  float* Pb1  = sm + 2 * 4096;     // buffer 1
  float* Ab1  = sm + 3 * 4096;     // buffer 1
  float* covs = sm + 4 * 4096;     // carry cov_sn
  float* T1   = sm + 5 * 4096;
  float* T2   = sm + 6 * 4096;     // Pn diff D, then J D J^T
  float* Sb   = sm + 7 * 4096;     // Pn -> Pn^-1
  float* Jb   = sm + 8 * 4096;     // J
  float* msn  = sm + 9 * 4096;     // 64 carry mean
  float* mp0  = msn + 64;          // 64
  float* dmv  = mp0 + 64;          // 64
  float* colk = dmv + 64;          // 64
  float* pivs = colk + 64;         // 1

  const int b = blockIdx.x;
  const int tid = threadIdx.x;
  const size_t last = (size_t)(T - 1) * B + b;

  // prologue: start DMA for t = T-2, then init carry from filter state
  {
    const size_t tb0 = (size_t)(T - 2) * B + b;
    float* Pn = ((T - 2) & 1) ? Pb1 : Pb0;
    float* An = ((T - 2) & 1) ? Ab1 : Ab0;
    async_tile_load4k(fcovs + tb0 * 4096, Pn, tid);
    async_tile_load4k(Asg + tb0 * 4096, An, tid);
  }
  for (int idx = tid; idx < 4096; idx += 256) covs[idx] = fcovs[last * 4096 + idx];
  if (tid < 64) {
    const float m = fmeans[last * 64 + tid];
    msn[tid] = m;
    out[last * 64 + tid] = m;                 // smooth_means[T-1] = f_means[T-1]
    mp0[tid] = mpred0g[(size_t)b * 64 + tid];
  }
  __syncthreads();

  for (int t = T - 2; t >= 0; --t) {
    const size_t tb = (size_t)t * B + b;
    float* Pb = (t & 1) ? Pb1 : Pb0;
    float* Ab = (t & 1) ? Ab1 : Ab0;

    wait_async_all();
    __syncthreads();
    if (t - 1 >= 0) {                          // prefetch next (earlier) step
      const size_t tbp = (size_t)(t - 1) * B + b;
      float* Pn = (t & 1) ? Pb0 : Pb1;
      float* An = (t & 1) ? Ab0 : Ab1;
      async_tile_load4k(fcovs + tbp * 4096, Pn, tid);
      async_tile_load4k(Asg + tbp * 4096, An, tid);
    }

    mm64<false, false>(Ab, Pb, T1, tid);      // T1 = A P
    __syncthreads();
    mm64<false, true>(T1, Ab, T2, tid);       // T2 = A P A^T
    __syncthreads();
    for (int idx = tid; idx < 4096; idx += 256) {   // Pn = sym(T2) + Q
      const int i = idx >> 6, j = idx & 63;
      Sb[idx] = 0.5f * (T2[idx] + T2[j * 64 + i]) + Qg[idx];
    }
    __syncthreads();
    for (int idx = tid; idx < 4096; idx += 256) T2[idx] = covs[idx] - Sb[idx];  // D
    __syncthreads();
    inv64_inplace(Sb, colk, pivs, tid);       // Sb = Pn^-1

    mm64<false, true>(Pb, Ab, T1, tid);       // T1 = P A^T
    __syncthreads();
    mm64<false, false>(T1, Sb, Jb, tid);      // J = P A^T Pn^-1
    __syncthreads();

    float nm = 0.f;
    if (tid < 64) dmv[tid] = msn[tid] - mp0[tid];
    __syncthreads();
    if (tid < 64) {                           // mean_t = m_t + J (mean_sn - mp0)
      float s = 0.f;
      for (int j = 0; j < 64; ++j) s += Jb[tid * 64 + j] * dmv[j];
      nm = fmeans[tb * 64 + tid] + s;
      out[tb * 64 + tid] = nm;
    }
    mm64<false, false>(Jb, T2, T1, tid);      // T1 = J D
    __syncthreads();
    mm64<false, true>(T1, Jb, T2, tid);       // T2 = J D J^T
    __syncthreads();
    for (int idx = tid; idx < 4096; idx += 256) {   // cov_sn = P + sym(T2)
      const int i = idx >> 6, j = idx & 63;
      covs[idx] = Pb[idx] + 0.5f * (T2[idx] + T2[j * 64 + i]);
    }
    if (tid < 64) msn[tid] = nm;
    __syncthreads();
  }
}

// ---------------------------------------------------------------------------
extern "C" void kernel_launch(void* const* d_in, const int* in_sizes, int n_in,
                              void* d_out, int out_size, void* d_ws, size_t ws_size,
                              hipStream_t stream) {
  (void)in_sizes; (void)n_in; (void)out_size; (void)ws_size;
  const int T = 128, B = 128;
  const float* as_ = (const float*)d_in[0];   // [T,B,64]
  const float* Ak  = (const float*)d_in[1];   // [32,64,64]
  const float* Ck  = (const float*)d_in[2];   // [32,64,64]
  const float* Lq  = (const float*)d_in[3];   // [64,64]
  const float* Lr  = (const float*)d_in[4];   // [64,64]
  const float* Wih = (const float*)d_in[5];   // [128,64]
  const float* Whh = (const float*)d_in[6];   // [128,32]
  const float* bih = (const float*)d_in[7];   // [128]
  const float* bhh = (const float*)d_in[8];   // [128]
  const float* im  = (const float*)d_in[9];   // [64]
  const float* ic  = (const float*)d_in[10];  // [64,64]
  float* out = (float*)d_out;                 // [T,B,64,1]

  float* ws = (float*)d_ws;
  size_t off = 0;
  float* Q      = ws + off; off += 4096;
  float* R      = ws + off; off += 4096;
  float* wbuf   = ws + off; off += (size_t)T * B * 32;
  float* Asb    = ws + off; off += (size_t)T * B * 4096;
  float* Csb    = ws + off; off += (size_t)T * B * 4096;
  float* fmean  = ws + off; off += (size_t)T * B * 64;
  float* fcov   = ws + off; off += (size_t)T * B * 4096;
  float* mpred0 = ws + off; off += (size_t)B * 64;

  qr_kernel<<<32, 256, 0, stream>>>(Lq, Lr, Q, R);
  lstm_kernel<<<B, 128, 0, stream>>>(as_, Wih, Whh, bih, bhh, wbuf, T, B);

  dim3 g2((T * B) / 64, 4096 / 64, 2);
  mix_gemm_kernel<<<g2, 256, 4096 * sizeof(float), stream>>>(wbuf, Ak, Ck, Asb, Csb);

  const size_t lds_f = (size_t)(8 * 4096 + 320) * sizeof(float);
  filter_kernel<<<B, 256, lds_f, stream>>>(as_, Asb, Csb, Q, R, im, ic,
                                           fmean, fcov, mpred0, T, B);

  const size_t lds_s = (size_t)(9 * 4096 + 320) * sizeof(float);
  smoother_kernel<<<B, 256, lds_s, stream>>>(Asb, Q, fmean, fcov, mpred0, out, T, B);
}